// PINN_5016521801754
// MI455X (gfx1250) — compile-verified
//
#include <hip/hip_runtime.h>

// CDNA5 / gfx1250 PINN beam-field kernel.
// Forward-mode jet propagation (value + d/dx + d2/dx2 + d3/dx3) through a
// 1->16->16->2 tanh MLP. The 16x16 layer-2 matvec (x4 derivative orders) is
// done in full fp32 precision on the matrix pipe via V_WMMA_F32_16X16X4_F32,
// 16 points per wave (wave32). tanh uses the gfx1250 hardware transcendental
// v_tanh_f32 where available (v_exp_f32 + v_rcp_f32 fallback).

typedef float v2f __attribute__((ext_vector_type(2)));
typedef float v8f __attribute__((ext_vector_type(8)));

#define LBEAM 1.0f
#define EA_C  1.0e4f
#define EI_C  1.0e2f

__device__ __forceinline__ float fast_tanhf(float a) {
#if __has_builtin(__builtin_amdgcn_tanhf)
    return __builtin_amdgcn_tanhf(a);          // v_tanh_f32 (gfx1250 TRANS op)
#else
    // stable tanh via hardware v_exp_f32 / v_rcp_f32: 1 - 2*rcp(exp(2a)+1)
    float e = __expf(2.0f * a);
    return 1.0f - 2.0f * __builtin_amdgcn_rcpf(e + 1.0f);
#endif
}

__global__ __launch_bounds__(256) void pinn_beam_wmma(
    const float* __restrict__ x,
    const float* __restrict__ W1, const float* __restrict__ b1,
    const float* __restrict__ W2, const float* __restrict__ b2,
    const float* __restrict__ W3, const float* __restrict__ b3,
    float* __restrict__ out, int n)
{
    const int lane = threadIdx.x & 31;
    const int half = lane >> 4;     // 0: channels {4c,4c+1} / D rows 0-7
    const int l16  = lane & 15;     // point within 16-point batch; W2 row for A

    // ---------- loop-invariant weight staging ----------
    // W2 in V_WMMA_F32_16X16X4_F32 A-matrix layout (M=row in lanes, K split
    // over VGPR pair + half-wave), 4 chunks of K=4.
    v2f Aw[4];
#pragma unroll
    for (int c = 0; c < 4; ++c) {
        Aw[c].x = W2[l16 * 16 + 4 * c + 2 * half + 0];
        Aw[c].y = W2[l16 * 16 + 4 * c + 2 * half + 1];
    }
    // layer-1 params for this lane's 8 B-matrix channels: ch = 4c + 2*half + j
    float w1c[8], b1c[8];
#pragma unroll
    for (int c = 0; c < 4; ++c)
#pragma unroll
        for (int j = 0; j < 2; ++j) {
            int ch = 4 * c + 2 * half + j;
            w1c[2 * c + j] = W1[ch];
            b1c[2 * c + j] = b1[ch];
        }
    // layer-2 bias and layer-3 rows for this lane's D rows: channel r + 8*half
    float b2c[8], w3u[8], w3w[8];
#pragma unroll
    for (int r = 0; r < 8; ++r) {
        int ch = r + 8 * half;
        b2c[r] = b2[ch];
        w3u[r] = W3[ch];          // W3[0][ch] -> output u
        w3w[r] = W3[16 + ch];     // W3[1][ch] -> output w
    }
    const float b30 = b3[0], b31 = b3[1];

    const int nBatch = n >> 4;
    const int gwave  = (int)(blockIdx.x * blockDim.x + threadIdx.x) >> 5;
    const int nwave  = (int)(gridDim.x * blockDim.x) >> 5;

    for (int batch = gwave; batch < nBatch; batch += nwave) {
        const int   pidx = batch * 16 + l16;
        const float xi   = x[pidx];

        // ---------- layer 1: tanh jets, 8 channels of this lane's point ----
        float h0[8], h1[8], h2[8], h3[8];
#pragma unroll
        for (int i = 0; i < 8; ++i) {
            float a1 = w1c[i] * (1.0f / LBEAM);   // da/dx (constant)
            float a0 = a1 * xi + b1c[i];
            float t  = fast_tanhf(a0);
            float s  = 1.0f - t * t;
            h0[i] = t;
            h1[i] = s * a1;
            h2[i] = -2.0f * t * s * a1 * a1;
            h3[i] = s * (4.0f * t * t - 2.0f * s) * a1 * a1 * a1;
        }

        // ---------- layer 2 pre-activation jets: A2_k = W2 @ H1_k ----------
        // four K=4 chunks x four derivative orders = 16 fp32 WMMAs
        v8f acc0 = {}, acc1 = {}, acc2 = {}, acc3 = {};
#pragma unroll
        for (int c = 0; c < 4; ++c) {
            v2f B0 = { h0[2 * c], h0[2 * c + 1] };
            v2f B1 = { h1[2 * c], h1[2 * c + 1] };
            v2f B2 = { h2[2 * c], h2[2 * c + 1] };
            v2f B3 = { h3[2 * c], h3[2 * c + 1] };
            acc0 = __builtin_amdgcn_wmma_f32_16x16x4_f32(false, Aw[c], false, B0,
                                                         (short)0, acc0, false, false);
            acc1 = __builtin_amdgcn_wmma_f32_16x16x4_f32(false, Aw[c], false, B1,
                                                         (short)0, acc1, false, false);
            acc2 = __builtin_amdgcn_wmma_f32_16x16x4_f32(false, Aw[c], false, B2,
                                                         (short)0, acc2, false, false);
            acc3 = __builtin_amdgcn_wmma_f32_16x16x4_f32(false, Aw[c], false, B3,
                                                         (short)0, acc3, false, false);
        }

        // ---------- layer-2 tanh jets + layer-3 partial sums ----------
        float p00 = 0.f, p01 = 0.f, p02 = 0.f, p03 = 0.f;  // u jets (partial)
        float p10 = 0.f, p11 = 0.f, p12 = 0.f, p13 = 0.f;  // w jets (partial)
#pragma unroll
        for (int r = 0; r < 8; ++r) {
            float A0 = acc0[r] + b2c[r];
            float A1 = acc1[r];
            float A2 = acc2[r];
            float A3 = acc3[r];
            float t  = fast_tanhf(A0);
            float s  = 1.0f - t * t;
            float g0 = t;
            float g1 = s * A1;
            float g2 = s * A2 - 2.0f * t * s * A1 * A1;
            float g3 = s * A3 - 6.0f * t * s * A1 * A2
                     + s * (4.0f * t * t - 2.0f * s) * A1 * A1 * A1;
            p00 += w3u[r] * g0; p01 += w3u[r] * g1;
            p02 += w3u[r] * g2; p03 += w3u[r] * g3;
            p10 += w3w[r] * g0; p11 += w3w[r] * g1;
            p12 += w3w[r] * g2; p13 += w3w[r] * g3;
        }
        // combine half-wave partials (channels 0-7 with 8-15): lane xor 16
        p00 += __shfl_xor(p00, 16, 32);
        p01 += __shfl_xor(p01, 16, 32);
        p02 += __shfl_xor(p02, 16, 32);
        p03 += __shfl_xor(p03, 16, 32);
        p10 += __shfl_xor(p10, 16, 32);
        p11 += __shfl_xor(p11, 16, 32);
        p12 += __shfl_xor(p12, 16, 32);
        p13 += __shfl_xor(p13, 16, 32);

        // ---------- assemble the 6 beam fields ----------
        float u    = p00 + b30;
        float ux   = p01;
        float wv   = p10 + b31;
        float wx   = p11;
        float wxx  = p12;
        float wxxx = p13;
        float Nax  = EA_C * (ux + 0.5f * wx * wx);
        float Mb   = -EI_C * wxx;
        float Qs   = -EI_C * wxxx + Nax * wx;

        // both half-waves hold all values: half 0 stores (u,w,wx),
        // half 1 stores (N,M,Q) -> all 32 lanes store, EXEC stays full.
        float f0 = half ? Nax : u;
        float f1 = half ? Mb  : wv;
        float f2 = half ? Qs  : wx;
        int base = (half ? 3 : 0) * n + pidx;
        out[base]         = f0;
        out[base + n]     = f1;
        out[base + 2 * n] = f2;
    }
}

extern "C" void kernel_launch(void* const* d_in, const int* in_sizes, int n_in,
                              void* d_out, int out_size, void* d_ws, size_t ws_size,
                              hipStream_t stream) {
    const float* x  = (const float*)d_in[0];
    const float* W1 = (const float*)d_in[1];
    const float* b1 = (const float*)d_in[2];
    const float* W2 = (const float*)d_in[3];
    const float* b2 = (const float*)d_in[4];
    const float* W3 = (const float*)d_in[5];
    const float* b3 = (const float*)d_in[6];
    float* out = (float*)d_out;
    const int n = in_sizes[0];

    const int nBatch = n >> 4;                       // 16 points per wave-batch
    int blocks = (nBatch * 32 + 255) / 256;          // 1 batch per wave ideally
    if (blocks > 2048) blocks = 2048;                // grid-stride beyond this
    if (blocks < 1) blocks = 1;
    pinn_beam_wmma<<<blocks, 256, 0, stream>>>(x, W1, b1, W2, b2, W3, b3, out, n);
}